// MultiVAEJumpTime_56006373539984
// MI455X (gfx1250) — compile-verified
//
#include <hip/hip_runtime.h>
#include <hip/hip_bf16.h>
#include <math.h>

typedef __attribute__((ext_vector_type(2))) float v2f;
typedef __attribute__((ext_vector_type(8))) float v8f;

#define NEXPERT 16
#define BATCH   1024
#define SCAP    1280        // max compacted slots: 1024 + 16*15 = 1264, padded
#define MTILES  8           // M tiles (of 16 rows) kept resident per block pass
#define KC      32          // K chunk staged in LDS
#define KQ      (KC / 4)    // K quads per chunk
#define LOG2PI_F 1.8378770664093453f

// ---------------------------------------------------------------------------
// helpers
// ---------------------------------------------------------------------------
__device__ __forceinline__ int compute_bin(float tv) {
    int idx = 0;
#pragma unroll
    for (int j = 1; j <= 15; j++) {
        float b;
        if (j < 7)       b = (float)j * (1.5f / 7.0f);
        else if (j < 13) b = 1.5f + (float)(j - 7) * (3.5f / 6.0f);
        else             b = 5.0f + (float)(j - 13) * (5.0f / 3.0f);
        idx += (tv > b) ? 1 : 0;
    }
    return idx;
}

__device__ __forceinline__ float silu_f(float x) {
    return x / (1.0f + __expf(-x));
}

// MADE masks, computed analytically.
__device__ __forceinline__ float made_mask(int type, int kin, int n, int parity) {
    if (type == 1) {
        if (kin >= 16) return 1.0f;                 // context inputs: degree -1
        int din = parity ? (15 - kin) : kin;
        return ((n % 15) >= din) ? 1.0f : 0.0f;
    } else if (type == 2) {
        return ((n % 15) >= (kin % 15)) ? 1.0f : 0.0f;
    } else {
        int o = n & 15;
        int dout = parity ? (15 - o) : o;
        return (dout > (kin % 15)) ? 1.0f : 0.0f;
    }
}

// ---------------------------------------------------------------------------
// setup: bin rows by expert, build compacted (16-padded) row map
// hdr[0..15]=cnt, hdr[16..31]=slot offset, hdr[32]=total slots
// ---------------------------------------------------------------------------
__global__ void setup_kernel(const float* __restrict__ t,
                             int* __restrict__ hdr, int* __restrict__ rowmap) {
    __shared__ int scnt[NEXPERT];
    __shared__ int soff[NEXPERT];
    __shared__ int sfill[NEXPERT];
    const int tid = threadIdx.x;
    if (tid < NEXPERT) { scnt[tid] = 0; sfill[tid] = 0; }
    __syncthreads();
    for (int b = tid; b < BATCH; b += blockDim.x)
        atomicAdd(&scnt[compute_bin(t[b])], 1);
    __syncthreads();
    if (tid == 0) {
        int off = 0;
        for (int e = 0; e < NEXPERT; e++) {
            soff[e] = off;
            off += ((scnt[e] + 15) >> 4) << 4;
        }
        hdr[32] = off;
    }
    __syncthreads();
    for (int b = tid; b < BATCH; b += blockDim.x) {
        int e = compute_bin(t[b]);
        int slot = atomicAdd(&sfill[e], 1);
        rowmap[soff[e] + slot] = b;
    }
    __syncthreads();
    if (tid < NEXPERT) { hdr[tid] = scnt[tid]; hdr[16 + tid] = soff[tid]; }
    for (int e = 0; e < NEXPERT; e++) {
        int pc = ((scnt[e] + 15) >> 4) << 4;
        for (int s = scnt[e] + tid; s < pc; s += blockDim.x)
            rowmap[soff[e] + s] = -1;
    }
}

// ---------------------------------------------------------------------------
// gather: build compacted input buffers per slot
// ---------------------------------------------------------------------------
__global__ void gather_kernel(const float* __restrict__ x_t,
                              const float* __restrict__ v_t,
                              const float* __restrict__ t,
                              const float* __restrict__ t_prev,
                              const int* __restrict__ hdr,
                              const int* __restrict__ rowmap,
                              float* __restrict__ Xg,
                              float* __restrict__ XDg,
                              float* __restrict__ Tg) {
    const int s = blockIdx.x;
    if (s >= hdr[32]) return;
    const int row = rowmap[s];
    const int tid = threadIdx.x;
    const float4 z4 = make_float4(0.f, 0.f, 0.f, 0.f);
    const float4* xs = (const float4*)(x_t + (size_t)(row < 0 ? 0 : row) * 1024);
    const float4* vs = (const float4*)(v_t + (size_t)(row < 0 ? 0 : row) * 1024);
    float4* xd = (float4*)(Xg + (size_t)s * 1024);
    float4* vd = (float4*)(XDg + (size_t)s * 1028);
    for (int i = tid; i < 256; i += blockDim.x) {
        xd[i] = (row >= 0) ? xs[i] : z4;
        vd[i] = (row >= 0) ? vs[i] : z4;
    }
    if (tid == 0) {
        XDg[(size_t)s * 1028 + 1024] = (row >= 0) ? t_prev[row] : 0.f;
        XDg[(size_t)s * 1028 + 1025] = 0.f;
        XDg[(size_t)s * 1028 + 1026] = 0.f;
        XDg[(size_t)s * 1028 + 1027] = 0.f;
        Tg[(size_t)s * 4 + 0] = (row >= 0) ? t[row] : 0.f;
        Tg[(size_t)s * 4 + 1] = 0.f;
        Tg[(size_t)s * 4 + 2] = 0.f;
        Tg[(size_t)s * 4 + 3] = 0.f;
    }
}

// ---------------------------------------------------------------------------
// Generic fused GEMM using V_WMMA_F32_16X16X4_F32.
//   C[slot][col0+n] = act( sum_k A[slot][k] * mask*W[e*egroup+kflow][k][n] + b[n] )
// act: 0=none, 1=silu, 2=relu.  maskType: 0=none,1=m1,2=m2,3=m3.
// ---------------------------------------------------------------------------
__global__ void gemm_wmma_kernel(const float* __restrict__ A, int lda, int Kpad,
                                 const float* __restrict__ W, int Kw, int N,
                                 const float* __restrict__ bias,
                                 float* __restrict__ C, int ldc, int col0,
                                 const int* __restrict__ hdr,
                                 int egroup, int kflow, int act, int maskType) {
    __shared__ float sA[MTILES * 16 * KC]; // 16 KB
    const int e = blockIdx.x;
    const int cnt = hdr[e];
    if (cnt == 0) return;
    const int slot0 = hdr[16 + e];
    const int tiles = (cnt + 15) >> 4;

    const int tid = threadIdx.x;
    const int wave = tid >> 5;
    const int lane = tid & 31;
    const int nwaves = blockDim.x >> 5;
    const int nbase = (blockIdx.y * nwaves + wave) * 16;
    const bool waveActive = (nbase < N);
    const bool fullN = (nbase + 16 <= N);
    const int ncol = nbase + (lane & 15);
    const int kk_half = (lane >> 4) << 1; // 0 for lanes 0-15, 2 for lanes 16-31
    const int m_lane = lane & 15;
    const int parity = kflow & 1;

    const size_t wstride = (size_t)Kw * (size_t)N;
    const float* We = W + (size_t)(e * egroup + kflow) * wstride;
    const float* be = bias ? (bias + (size_t)(e * egroup + kflow) * (size_t)N) : nullptr;

    for (int mchunk = blockIdx.z; mchunk * MTILES < tiles; mchunk += gridDim.z) {
        const int mt0 = mchunk * MTILES;
        const int tcount = min(MTILES, tiles - mt0);
        const bool fastwave = fullN && (maskType == 0) && (tcount == MTILES);

        v8f acc[MTILES];
        const v8f vzero = {0.f, 0.f, 0.f, 0.f, 0.f, 0.f, 0.f, 0.f};
#pragma unroll
        for (int i = 0; i < MTILES; i++) acc[i] = vzero;

        for (int kc = 0; kc < Kpad; kc += KC) {
            __syncthreads();
            // ---- stage A tiles (tcount x 16 rows x KC cols) into LDS ----
            if (kc + KC <= Kpad && tcount == MTILES) {
                for (int i4 = tid; i4 < MTILES * 16 * KQ; i4 += blockDim.x) {
                    const int c4 = i4 % KQ;
                    const int rr = i4 / KQ;
                    const float* src =
                        A + (size_t)(slot0 + mt0 * 16 + rr) * (size_t)lda + kc + c4 * 4;
                    *(float4*)&sA[rr * KC + c4 * 4] = *(const float4*)src;
                }
            } else {
                const int kcn = min(KC, Kpad - kc);
                const int nquads = tcount * 16 * KQ;
                for (int i4 = tid; i4 < nquads; i4 += blockDim.x) {
                    const int c4 = i4 % KQ;
                    const int rr = i4 / KQ;
                    const int col = c4 * 4;
                    float4 v = make_float4(0.f, 0.f, 0.f, 0.f);
                    if (col < kcn) {
                        const float* src =
                            A + (size_t)(slot0 + mt0 * 16 + rr) * (size_t)lda + kc + col;
                        v = *(const float4*)src;
                    }
                    *(float4*)&sA[rr * KC + col] = v;
                }
            }
            __syncthreads();
            if (!waveActive) continue;

            if (fastwave && (kc + KC <= Kpad) && (kc + KC <= Kw)) {
                // ---- hot path: preload whole W chunk (16 independent loads),
                //      prefetch next chunk, then fully unrolled ds+wmma stream ----
                float wreg[2 * KQ];
#pragma unroll
                for (int k4i = 0; k4i < KQ; k4i++) {
                    const int kg = kc + k4i * 4 + kk_half;
                    wreg[2 * k4i]     = We[(size_t)kg * N + ncol];
                    wreg[2 * k4i + 1] = We[(size_t)(kg + 1) * N + ncol];
                }
                if (kc + 2 * KC <= Kw) {
#pragma unroll
                    for (int k4i = 0; k4i < KQ; k4i++) {
                        const int kg = kc + KC + k4i * 4 + kk_half;
                        __builtin_prefetch(&We[(size_t)kg * N + ncol], 0, 1);
                        __builtin_prefetch(&We[(size_t)(kg + 1) * N + ncol], 0, 1);
                    }
                }
#pragma unroll
                for (int k4i = 0; k4i < KQ; k4i++) {
                    v2f bvec;
                    bvec[0] = wreg[2 * k4i];
                    bvec[1] = wreg[2 * k4i + 1];
                    v2f a[MTILES];
#pragma unroll
                    for (int mt = 0; mt < MTILES; mt++) {
                        const float* ap = &sA[(mt * 16 + m_lane) * KC + k4i * 4 + kk_half];
                        a[mt][0] = ap[0];
                        a[mt][1] = ap[1];
                    }
#pragma unroll
                    for (int mt = 0; mt < MTILES; mt++)
                        acc[mt] = __builtin_amdgcn_wmma_f32_16x16x4_f32(
                            false, a[mt], false, bvec, (short)0, acc[mt], false, false);
                }
            } else {
                // ---- generic guarded path ----
                const int kend = min(KC, Kpad - kc);
                for (int k4 = 0; k4 < kend; k4 += 4) {
                    const int kg = kc + k4 + kk_half;
                    float w0 = 0.f, w1 = 0.f;
                    if (ncol < N) {
                        if (kg < Kw)     w0 = We[(size_t)kg * N + ncol];
                        if (kg + 1 < Kw) w1 = We[(size_t)(kg + 1) * N + ncol];
                        if (maskType) {
                            w0 *= made_mask(maskType, kg, ncol, parity);
                            w1 *= made_mask(maskType, kg + 1, ncol, parity);
                        }
                    }
                    v2f bvec; bvec[0] = w0; bvec[1] = w1;
#pragma unroll
                    for (int mt = 0; mt < MTILES; mt++) {
                        if (mt >= tcount) break;
                        const float* ap = &sA[(mt * 16 + m_lane) * KC + k4 + kk_half];
                        v2f avec; avec[0] = ap[0]; avec[1] = ap[1];
                        acc[mt] = __builtin_amdgcn_wmma_f32_16x16x4_f32(
                            false, avec, false, bvec, (short)0, acc[mt], false, false);
                    }
                }
            }
        }

        // ---- epilogue: bias + activation + store ----
        if (waveActive && ncol < N) {
            const float bv = be ? be[ncol] : 0.f;
#pragma unroll
            for (int mt = 0; mt < MTILES; mt++) {
                if (mt >= tcount) break;
#pragma unroll
                for (int r = 0; r < 8; r++) {
                    const int m = r + ((lane >> 4) << 3);
                    float x = acc[mt][r] + bv;
                    if (act == 1)      x = silu_f(x);
                    else if (act == 2) x = fmaxf(x, 0.f);
                    const int slot = slot0 + (mt0 + mt) * 16 + m;
                    C[(size_t)slot * (size_t)ldc + col0 + ncol] = x;
                }
            }
        }
    }
}

// ---------------------------------------------------------------------------
// sampling: z = mu_q + exp(ls_q)*eps ; log_q ; init ladj
// ---------------------------------------------------------------------------
__global__ void sample_kernel(const float* __restrict__ eps,
                              const int* __restrict__ hdr,
                              const int* __restrict__ rowmap,
                              const float* __restrict__ PHI,
                              float* __restrict__ FIN,
                              float* __restrict__ logq,
                              float* __restrict__ ladj) {
    const int s = blockIdx.x * blockDim.x + threadIdx.x;
    if (s >= hdr[32]) return;
    const int row = rowmap[s];
    float lq = 0.f;
#pragma unroll
    for (int j = 0; j < 16; j++) {
        const float mu = PHI[(size_t)s * 32 + j];
        const float ls = PHI[(size_t)s * 32 + 16 + j];
        const float ep = (row >= 0) ? eps[(size_t)row * 16 + j] : 0.f;
        FIN[(size_t)s * 272 + j] = mu + __expf(ls) * ep;
        lq += -0.5f * ep * ep - ls - 0.5f * LOG2PI_F;
    }
    logq[s] = lq;
    ladj[s] = 0.f;
}

// ---------------------------------------------------------------------------
// log_px: wave-per-slot reduction over 1025 dims
// ---------------------------------------------------------------------------
__global__ void logpx_kernel(const int* __restrict__ hdr,
                             const float* __restrict__ XDg,
                             const float* __restrict__ PD,
                             float* __restrict__ logpx) {
    const int wave = threadIdx.x >> 5;
    const int lane = threadIdx.x & 31;
    const int s = blockIdx.x * (blockDim.x >> 5) + wave;
    if (s >= hdr[32]) return;
    float acc = 0.f;
    for (int i = lane; i < 1025; i += 32) {
        const float xv = XDg[(size_t)s * 1028 + i];
        const float mu = PD[(size_t)s * 2050 + i];
        const float ls = PD[(size_t)s * 2050 + 1025 + i];
        const float d = (xv - mu) * __expf(-ls);
        acc += -0.5f * d * d - ls - 0.5f * LOG2PI_F;
    }
#pragma unroll
    for (int m = 16; m >= 1; m >>= 1) acc += __shfl_xor(acc, m, 32);
    if (lane == 0) logpx[s] = acc;
}

// ---------------------------------------------------------------------------
// flow update: y = y*exp(ls) + sh ; ladj += sum(ls)
// ---------------------------------------------------------------------------
__global__ void flow_update_kernel(const int* __restrict__ hdr,
                                   const float* __restrict__ PH,
                                   float* __restrict__ FIN,
                                   float* __restrict__ ladj) {
    const int s = blockIdx.x * blockDim.x + threadIdx.x;
    if (s >= hdr[32]) return;
    float la = 0.f;
#pragma unroll
    for (int j = 0; j < 16; j++) {
        const float sh = PH[(size_t)s * 32 + j];
        const float ls = PH[(size_t)s * 32 + 16 + j];
        la += ls;
        const float y = FIN[(size_t)s * 272 + j];
        FIN[(size_t)s * 272 + j] = y * __expf(ls) + sh;
    }
    ladj[s] += la;
}

// ---------------------------------------------------------------------------
// final: elbo = log_px + log_prior - log_q scattered to out[row]
// ---------------------------------------------------------------------------
__global__ void final_kernel(const int* __restrict__ hdr,
                             const int* __restrict__ rowmap,
                             const float* __restrict__ FIN,
                             const float* __restrict__ logq,
                             const float* __restrict__ logpx,
                             const float* __restrict__ ladj,
                             float* __restrict__ out) {
    const int s = blockIdx.x * blockDim.x + threadIdx.x;
    if (s >= hdr[32]) return;
    const int row = rowmap[s];
    if (row < 0) return;
    float lp = 0.f;
#pragma unroll
    for (int j = 0; j < 16; j++) {
        const float y = FIN[(size_t)s * 272 + j];
        lp += -0.5f * y * y - 0.5f * LOG2PI_F;
    }
    lp += ladj[s];
    out[row] = logpx[s] + lp - logq[s];
}

// ---------------------------------------------------------------------------
// host launcher
// ---------------------------------------------------------------------------
extern "C" void kernel_launch(void* const* d_in, const int* in_sizes, int n_in,
                              void* d_out, int out_size, void* d_ws, size_t ws_size,
                              hipStream_t stream) {
    const float* x_t    = (const float*)d_in[0];
    const float* v_t    = (const float*)d_in[1];
    const float* t      = (const float*)d_in[2];
    const float* t_prev = (const float*)d_in[3];
    const float* eps    = (const float*)d_in[4];
    const float* xW1 = (const float*)d_in[5];   const float* xb1 = (const float*)d_in[6];
    const float* xW2 = (const float*)d_in[7];   const float* xb2 = (const float*)d_in[8];
    const float* xW3 = (const float*)d_in[9];   const float* xb3 = (const float*)d_in[10];
    const float* tW1 = (const float*)d_in[11];  const float* tb1 = (const float*)d_in[12];
    const float* tW2 = (const float*)d_in[13];  const float* tb2 = (const float*)d_in[14];
    const float* tW3 = (const float*)d_in[15];  const float* tb3 = (const float*)d_in[16];
    const float* eW1 = (const float*)d_in[17];  const float* eb1 = (const float*)d_in[18];
    const float* eW2 = (const float*)d_in[19];  const float* eb2 = (const float*)d_in[20];
    const float* eW3 = (const float*)d_in[21];  const float* eb3 = (const float*)d_in[22];
    const float* dW1 = (const float*)d_in[23];  const float* db1 = (const float*)d_in[24];
    const float* dW2 = (const float*)d_in[25];  const float* db2 = (const float*)d_in[26];
    const float* dW3 = (const float*)d_in[27];  const float* db3 = (const float*)d_in[28];
    const float* mW1 = (const float*)d_in[29];  const float* mb1 = (const float*)d_in[30];
    const float* mW2 = (const float*)d_in[31];  const float* mb2 = (const float*)d_in[32];
    const float* mW3 = (const float*)d_in[33];  const float* mb3 = (const float*)d_in[34];
    float* out = (float*)d_out;

    // workspace layout
    int* hdr    = (int*)d_ws;              // 64 ints
    int* rowmap = hdr + 64;                // SCAP ints
    float* fbase = (float*)((char*)d_ws + 8192);
    size_t fo = 0;
    auto alloc = [&](size_t n) { float* p = fbase + fo; fo += (n + 15) & ~(size_t)15; return p; };
    float* Xg   = alloc((size_t)SCAP * 1024);
    float* XDg  = alloc((size_t)SCAP * 1028);
    float* Tg   = alloc((size_t)SCAP * 4);
    float* H1   = alloc((size_t)SCAP * 1280);
    float* H2   = alloc((size_t)SCAP * 1280);
    float* FIN  = alloc((size_t)SCAP * 272);
    float* PHI  = alloc((size_t)SCAP * 32);
    float* PD   = alloc((size_t)SCAP * 2050);
    float* PH   = alloc((size_t)SCAP * 32);
    float* logq = alloc(SCAP);
    float* lpx  = alloc(SCAP);
    float* ladj = alloc(SCAP);
    (void)ws_size; (void)in_sizes; (void)n_in; (void)out_size;

    setup_kernel<<<1, 256, 0, stream>>>(t, hdr, rowmap);
    gather_kernel<<<SCAP, 256, 0, stream>>>(x_t, v_t, t, t_prev, hdr, rowmap, Xg, XDg, Tg);

    auto gemm = [&](const float* A, int lda, int Kpad, const float* W, int Kw, int N,
                    const float* bias, float* C, int ldc, int col0,
                    int egroup, int kflow, int act, int maskType, int threads) {
        const int waves = threads / 32;
        dim3 grid(NEXPERT, (N + 16 * waves - 1) / (16 * waves), 2);
        gemm_wmma_kernel<<<grid, threads, 0, stream>>>(A, lda, Kpad, W, Kw, N, bias,
                                                       C, ldc, col0, hdr,
                                                       egroup, kflow, act, maskType);
    };

    // x encoder: silu
    gemm(Xg, 1024, 1024, xW1, 1024, 1280, xb1, H1, 1280, 0,   1, 0, 1, 0, 512);
    gemm(H1, 1280, 1280, xW2, 1280, 1280, xb2, H2, 1280, 0,   1, 0, 1, 0, 512);
    gemm(H2, 1280, 1280, xW3, 1280, 128,  xb3, FIN, 272, 16,  1, 0, 1, 0, 256);
    // t encoder: silu
    gemm(Tg, 4, 4,       tW1, 1,    256,  tb1, H1, 1280, 0,   1, 0, 1, 0, 512);
    gemm(H1, 1280, 256,  tW2, 256,  256,  tb2, H2, 1280, 0,   1, 0, 1, 0, 512);
    gemm(H2, 1280, 256,  tW3, 256,  128,  tb3, FIN, 272, 144, 1, 0, 1, 0, 256);
    // encoder q(z|x): relu, relu, linear
    gemm(XDg, 1028, 1028, eW1, 1025, 1280, eb1, H1, 1280, 0,  1, 0, 2, 0, 512);
    gemm(H1, 1280, 1280,  eW2, 1280, 1280, eb2, H2, 1280, 0,  1, 0, 2, 0, 512);
    gemm(H2, 1280, 1280,  eW3, 1280, 32,   eb3, PHI, 32, 0,   1, 0, 0, 0, 64);
    sample_kernel<<<(SCAP + 255) / 256, 256, 0, stream>>>(eps, hdr, rowmap, PHI, FIN, logq, ladj);
    // decoder p(x|z): relu, relu, linear
    gemm(FIN, 272, 16,   dW1, 16,   1280, db1, H1, 1280, 0,   1, 0, 2, 0, 512);
    gemm(H1, 1280, 1280, dW2, 1280, 1280, db2, H2, 1280, 0,   1, 0, 2, 0, 512);
    gemm(H2, 1280, 1280, dW3, 1280, 2050, db3, PD, 2050, 0,   1, 0, 0, 0, 512);
    logpx_kernel<<<(SCAP + 7) / 8, 256, 0, stream>>>(hdr, XDg, PD, lpx);
    // 3 MADE flows
    for (int k = 0; k < 3; k++) {
        gemm(FIN, 272, 272, mW1, 272, 256, mb1, H1, 1280, 0,  3, k, 2, 1, 512);
        gemm(H1, 1280, 256, mW2, 256, 256, mb2, H2, 1280, 0,  3, k, 2, 2, 512);
        gemm(H2, 1280, 256, mW3, 256, 32,  mb3, PH, 32, 0,    3, k, 0, 3, 64);
        flow_update_kernel<<<(SCAP + 255) / 256, 256, 0, stream>>>(hdr, PH, FIN, ladj);
    }
    final_kernel<<<(SCAP + 255) / 256, 256, 0, stream>>>(hdr, rowmap, FIN, logq, lpx, ladj, out);
}